// ROIPooler_25099788877849
// MI455X (gfx1250) — compile-verified
//
#include <hip/hip_runtime.h>

typedef __attribute__((ext_vector_type(2))) float v2f;
typedef __attribute__((ext_vector_type(8))) float v8f;
typedef __attribute__((ext_vector_type(4), aligned(4))) float v4fu;  // dword-aligned b128

#define NROI  512
#define RPB   256
#define NCH   256
#define OUTS  7
#define MAXP  132   // max padded footprint per axis (worst aspect: 512*0.25 + 2 -> 130)
#define TX    32    // x-tile for stage-1 -> stage-2 handoff

// torchvision ROIAlign (aligned=false) bilinear index prep, weights folded with
// the 0.5-per-axis share of the 1/4 sample mean and the validity mask.
__device__ __forceinline__ void prep(float c, float lim,
                                     int& lo, int& hi, float& wl, float& wh) {
    bool  valid = (c > -1.0f) && (c < lim);
    float cc    = fmaxf(c, 0.0f);
    float low   = floorf(cc);
    float cap   = lim - 1.0f;
    bool  edge  = (low >= cap);
    low         = fminf(low, cap);
    float high  = fminf(low + 1.0f, cap);
    cc          = edge ? low : cc;
    float f     = cc - low;
    lo = (int)low;
    hi = (int)high;
    float v = valid ? 0.5f : 0.0f;
    wl = (1.0f - f) * v;
    wh = f * v;
}

__device__ __forceinline__ v8f wmma4(v2f a, v2f b, v8f c) {
    // D(16x16) += A(16x4) * B(4x16), fp32
    return __builtin_amdgcn_wmma_f32_16x16x4_f32(false, a, false, b,
                                                 (short)0, c, false, false);
}

__global__ __launch_bounds__(32) void roialign_sep_wmma(
    const float* __restrict__ f0, const float* __restrict__ f1,
    const float* __restrict__ f2, const float* __restrict__ f3,
    const float* __restrict__ boxes, float* __restrict__ out)
{
    __shared__ float Wy[MAXP][16];        // [k = yr-ymin][m = by], rows 7..15 zero
    __shared__ float Wx[MAXP][16];        // [k = xr-xmin][m = bx]
    __shared__ float M1[OUTS][TX][16];    // stage-1 result: [by][xr in tile][c]

    const int blk   = blockIdx.x;
    const int roi   = blk >> 4;           // 512 ROIs
    const int chunk = blk & 15;           // 16 channel-chunks of 16
    const int lane  = (int)threadIdx.x;   // wave32

    const int batch = roi / RPB;

    // ---- per-ROI parameters (computed redundantly by all lanes) ----
    const float x1b = boxes[roi * 4 + 0];
    const float y1b = boxes[roi * 4 + 1];
    const float x2b = boxes[roi * 4 + 2];
    const float y2b = boxes[roi * 4 + 3];

    float area = (x2b - x1b) * (y2b - y1b);
    float lf   = floorf(4.0f + log2f(sqrtf(area) / 224.0f + 1e-8f));
    int   lvl  = (int)(fminf(fmaxf(lf, 2.0f), 5.0f)) - 2;

    const float scales[4] = {0.25f, 0.125f, 0.0625f, 0.03125f};
    const int   dims[4]   = {200, 100, 50, 25};
    const float sc  = scales[lvl];
    const int   HW  = dims[lvl];
    const float lim = (float)HW;
    const float* fsel = (lvl == 0) ? f0 : (lvl == 1) ? f1 : (lvl == 2) ? f2 : f3;

    const float x1 = x1b * sc, y1 = y1b * sc;
    const float x2 = x2b * sc, y2 = y2b * sc;
    const float roi_w = fmaxf(x2 - x1, 1.0f);
    const float roi_h = fmaxf(y2 - y1, 1.0f);
    const float bw = roi_w / 7.0f;
    const float bh = roi_h / 7.0f;

    // footprint bounds: sample coords are monotone in s, so endpoints suffice
    int lo, hi; float wl, wh;
    prep(y1 + bh * 0.25f, lim, lo, hi, wl, wh);  const int ymin = lo;
    prep(y1 + bh * 6.75f, lim, lo, hi, wl, wh);  const int ymax = hi;
    prep(x1 + bw * 0.25f, lim, lo, hi, wl, wh);  const int xmin = lo;
    prep(x1 + bw * 6.75f, lim, lo, hi, wl, wh);  const int xmax = hi;

    int Py4 = ((ymax - ymin + 1) + 3) & ~3;  if (Py4 > MAXP) Py4 = MAXP;
    int Px4 = ((xmax - xmin + 1) + 3) & ~3;  if (Px4 > MAXP) Px4 = MAXP;

    // ---- build separable weight matrices in LDS ----
    for (int i = lane; i < Py4 * 16; i += 32) ((float*)Wy)[i] = 0.0f;
    for (int i = lane; i < Px4 * 16; i += 32) ((float*)Wx)[i] = 0.0f;
    __syncthreads();
    if (lane == 0) {
        #pragma unroll
        for (int s = 0; s < 14; ++s) {
            float g = (float)(s >> 1) + ((float)(s & 1) + 0.5f) * 0.5f;
            int mm = s >> 1;
            prep(y1 + bh * g, lim, lo, hi, wl, wh);
            Wy[lo - ymin][mm] += wl;
            Wy[hi - ymin][mm] += wh;
            prep(x1 + bw * g, lim, lo, hi, wl, wh);
            Wx[lo - xmin][mm] += wl;
            Wx[hi - xmin][mm] += wh;
        }
    }
    __syncthreads();

    // ---- fragment lane roles (16x16x4 f32 layout) ----
    const int m      = lane & 15;           // A-matrix row (bin index, 7..15 padded)
    const int koff   = (lane >> 4) * 2;     // K sub-offset: lanes 0-15 -> K0/K1, 16-31 -> K2/K3
    const int clocal = lane & 15;           // B-matrix column (channel in chunk)
    const float* fc = fsel +
        ((size_t)batch * NCH + (size_t)(chunk * 16 + clocal)) *
        ((size_t)HW * (size_t)HW);

    v8f acc[OUTS];
    #pragma unroll
    for (int i = 0; i < OUTS; ++i) acc[i] = v8f{};

    for (int xt = 0; xt < Px4; xt += TX) {
        const int tw = (Px4 - xt < TX) ? (Px4 - xt) : TX;   // multiple of 4

        // ---- stage 1: M1[by][xi][c] = sum_yr Wy[by][yr] * V[yr][x][c] ----
        for (int xi = 0; xi < tw; xi += 4) {
            const int xbase = xmin + xt + xi;
            v8f d0 = v8f{}, d1 = v8f{}, d2 = v8f{}, d3 = v8f{};

            if (xbase + 4 <= HW) {
                // vector path: b128 per row, software-pipelined so next step's
                // loads are in flight under the current 4-WMMA burst.
                const float* fcx = fc + xbase;           // 64-bit add once
                int y0 = ymin + koff;     if (y0 > ymax) y0 = ymax;
                int y1r = ymin + koff + 1; if (y1r > ymax) y1r = ymax;
                v4fu r0 = *(const v4fu*)(fcx + y0 * HW);   // 32-bit row offsets
                v4fu r1 = *(const v4fu*)(fcx + y1r * HW);

                int ks = 0;
                for (; ks + 4 < Py4; ks += 4) {
                    v2f a;
                    a.x = Wy[ks + koff][m];
                    a.y = Wy[ks + koff + 1][m];
                    v4fu c0 = r0, c1 = r1;
                    int yn0 = ymin + ks + 4 + koff;     if (yn0 > ymax) yn0 = ymax;
                    int yn1 = ymin + ks + 4 + koff + 1; if (yn1 > ymax) yn1 = ymax;
                    r0 = *(const v4fu*)(fcx + yn0 * HW);
                    r1 = *(const v4fu*)(fcx + yn1 * HW);
                    v2f b0; b0.x = c0.x; b0.y = c1.x;
                    v2f b1; b1.x = c0.y; b1.y = c1.y;
                    v2f b2; b2.x = c0.z; b2.y = c1.z;
                    v2f b3; b3.x = c0.w; b3.y = c1.w;
                    d0 = wmma4(a, b0, d0);
                    d1 = wmma4(a, b1, d1);
                    d2 = wmma4(a, b2, d2);
                    d3 = wmma4(a, b3, d3);
                }
                // epilogue: consume the last preloaded pair
                {
                    v2f a;
                    a.x = Wy[ks + koff][m];
                    a.y = Wy[ks + koff + 1][m];
                    v2f b0; b0.x = r0.x; b0.y = r1.x;
                    v2f b1; b1.x = r0.y; b1.y = r1.y;
                    v2f b2; b2.x = r0.z; b2.y = r1.z;
                    v2f b3; b3.x = r0.w; b3.y = r1.w;
                    d0 = wmma4(a, b0, d0);
                    d1 = wmma4(a, b1, d1);
                    d2 = wmma4(a, b2, d2);
                    d3 = wmma4(a, b3, d3);
                }
            } else {
                // edge path (group crosses end of feature row): clamped scalars
                for (int ks = 0; ks < Py4; ks += 4) {
                    const int k0 = ks + koff;
                    v2f a;
                    a.x = Wy[k0][m];
                    a.y = Wy[k0 + 1][m];
                    int yr0 = ymin + k0;     if (yr0 > ymax) yr0 = ymax;
                    int yr1 = ymin + k0 + 1; if (yr1 > ymax) yr1 = ymax;
                    v4fu r0, r1;
                    #pragma unroll
                    for (int j = 0; j < 4; ++j) {
                        int xc = xbase + j; if (xc > HW - 1) xc = HW - 1;
                        r0[j] = fc[yr0 * HW + xc];
                        r1[j] = fc[yr1 * HW + xc];
                    }
                    v2f b0; b0.x = r0.x; b0.y = r1.x;
                    v2f b1; b1.x = r0.y; b1.y = r1.y;
                    v2f b2; b2.x = r0.z; b2.y = r1.z;
                    v2f b3; b3.x = r0.w; b3.y = r1.w;
                    d0 = wmma4(a, b0, d0);
                    d1 = wmma4(a, b1, d1);
                    d2 = wmma4(a, b2, d2);
                    d3 = wmma4(a, b3, d3);
                }
            }

            if (lane < 16) {                    // rows 0..6 live in lanes 0..15
                #pragma unroll
                for (int r = 0; r < OUTS; ++r) {
                    M1[r][xi + 0][lane] = d0[r];
                    M1[r][xi + 1][lane] = d1[r];
                    M1[r][xi + 2][lane] = d2[r];
                    M1[r][xi + 3][lane] = d3[r];
                }
            }
        }
        __syncthreads();

        // ---- stage 2: acc[by][bx][c] += sum_xr Wx[bx][xr] * M1[by][xr][c] ----
        // ks outer: A-fragment loaded once, 7 independent WMMAs per K-step.
        for (int ks = 0; ks < tw; ks += 4) {
            const int k0 = ks + koff;
            v2f a;
            a.x = Wx[xt + k0][m];
            a.y = Wx[xt + k0 + 1][m];
            #pragma unroll
            for (int by = 0; by < OUTS; ++by) {
                v2f b;
                b.x = M1[by][k0][clocal];
                b.y = M1[by][k0 + 1][clocal];
                acc[by] = wmma4(a, b, acc[by]);
            }
        }
        __syncthreads();
    }

    // ---- writeback: lanes 0..15 hold D rows 0..7 (bx) for channel clocal ----
    if (lane < 16) {
        float* op = out + ((size_t)roi * NCH + (size_t)(chunk * 16 + lane)) * 49;
        #pragma unroll
        for (int by = 0; by < OUTS; ++by) {
            #pragma unroll
            for (int bx = 0; bx < OUTS; ++bx) {
                op[by * 7 + bx] = acc[by][bx];
            }
        }
    }
}

extern "C" void kernel_launch(void* const* d_in, const int* in_sizes, int n_in,
                              void* d_out, int out_size, void* d_ws, size_t ws_size,
                              hipStream_t stream) {
    (void)in_sizes; (void)n_in; (void)d_ws; (void)ws_size; (void)out_size;
    const float* f0    = (const float*)d_in[0];
    const float* f1    = (const float*)d_in[1];
    const float* f2    = (const float*)d_in[2];
    const float* f3    = (const float*)d_in[3];
    const float* boxes = (const float*)d_in[4];
    float* out = (float*)d_out;

    dim3 grid(NROI * 16);   // (ROI, 16-channel chunk) per single-wave workgroup
    dim3 block(32);
    hipLaunchKernelGGL(roialign_sep_wmma, grid, block, 0, stream,
                       f0, f1, f2, f3, boxes, out);
}